// Quantizer_25159918420223
// MI455X (gfx1250) — compile-verified
//
#include <hip/hip_runtime.h>
#include <cstdint>
#include <cstddef>

// int4 symmetric quantizer, per-row scale = fp16(max|row|/7).
// x: [ROWS, COLS] fp32.  d_out = [ROWS*COLS floats of q] ++ [ROWS floats of scale].
#define COLS 4096
#define THREADS 256
#define V4_PER_THREAD 4   // 4 x float4 = 16 floats/thread; 256 threads * 16 = 4096
#define NWAVES (THREADS / 32)

__global__ __launch_bounds__(THREADS)
void Quantizer_25159918420223_kernel(const float* __restrict__ x,
                                     float* __restrict__ qout,
                                     float* __restrict__ sout)
{
    __shared__ __align__(16) float tile[COLS];   // one 16 KB row staged in LDS
    __shared__ float wred[NWAVES];
    __shared__ float s_scale;
    __shared__ float s_inv;

    const int row = blockIdx.x;
    const size_t rbase = (size_t)row * COLS;
    const float* __restrict__ grow = x + rbase;

    // ---- Stage the row into LDS via CDNA5 async global->LDS b128 copies ----
    // Per-lane: 4 async b128 transfers; fully coalesced (lanes cover contiguous
    // 512B per instruction). Tracked with ASYNCcnt, not LOADcnt.
#pragma unroll
    for (int i = 0; i < V4_PER_THREAD; ++i) {
        const int e = (i * THREADS + (int)threadIdx.x) * 4;     // float index
        const uint32_t lds_addr = (uint32_t)(uintptr_t)(&tile[e]); // low 32 bits = LDS offset
        const uint64_t gaddr    = (uint64_t)(uintptr_t)(grow + e);
        asm volatile("global_load_async_to_lds_b128 %0, %1, off"
                     :: "v"(lds_addr), "v"(gaddr)
                     : "memory");
    }
    asm volatile("s_wait_asynccnt 0x0" ::: "memory");
    __syncthreads();   // make every wave's staged portion visible block-wide

    // ---- Pass 1: pull 16 floats/thread from LDS, reduce max|x| ----
    float4 v[V4_PER_THREAD];
    float m = 0.0f;
#pragma unroll
    for (int i = 0; i < V4_PER_THREAD; ++i) {
        const int e = (i * THREADS + (int)threadIdx.x) * 4;
        v[i] = *(const float4*)&tile[e];                       // ds_load_b128
        m = fmaxf(m, fmaxf(fmaxf(__builtin_fabsf(v[i].x), __builtin_fabsf(v[i].y)),
                           fmaxf(__builtin_fabsf(v[i].z), __builtin_fabsf(v[i].w))));
    }

    // wave32 butterfly reduction (5 steps)
#pragma unroll
    for (int off = 16; off > 0; off >>= 1)
        m = fmaxf(m, __shfl_xor(m, off, 32));

    const int lane = (int)threadIdx.x & 31;
    const int wv   = (int)threadIdx.x >> 5;
    if (lane == 0) wred[wv] = m;
    __syncthreads();

    if (threadIdx.x == 0) {
        float rm = wred[0];
#pragma unroll
        for (int i = 1; i < NWAVES; ++i) rm = fmaxf(rm, wred[i]);
        // Faithful to reference: (max|row| / 7) rounded to fp16, clip ratio 1.0
        const _Float16 h = (_Float16)(rm / 7.0f);
        const float sc = (float)h;
        s_scale = sc;
        s_inv = 1.0f / sc;
        sout[row] = sc;                // scale output (fp16 value, stored as f32)
    }
    __syncthreads();

    const float inv = s_inv;
    float* __restrict__ qrow = qout + rbase;

    // ---- Pass 2: quantize from registers, b128 stores ----
#pragma unroll
    for (int i = 0; i < V4_PER_THREAD; ++i) {
        const int e = (i * THREADS + (int)threadIdx.x) * 4;
        float4 q;
        q.x = fminf(fmaxf(__builtin_rintf(v[i].x * inv), -8.0f), 7.0f);
        q.y = fminf(fmaxf(__builtin_rintf(v[i].y * inv), -8.0f), 7.0f);
        q.z = fminf(fmaxf(__builtin_rintf(v[i].z * inv), -8.0f), 7.0f);
        q.w = fminf(fmaxf(__builtin_rintf(v[i].w * inv), -8.0f), 7.0f);
        *(float4*)&qrow[e] = q;                                // global_store_b128
    }
}

extern "C" void kernel_launch(void* const* d_in, const int* in_sizes, int n_in,
                              void* d_out, int out_size, void* d_ws, size_t ws_size,
                              hipStream_t stream)
{
    (void)n_in; (void)out_size; (void)d_ws; (void)ws_size;
    const float* x = (const float*)d_in[0];
    const int rows = in_sizes[0] / COLS;          // 16384
    float* qout = (float*)d_out;                  // rows*COLS quantized values
    float* sout = (float*)d_out + (size_t)rows * COLS; // rows scales
    Quantizer_25159918420223_kernel<<<rows, THREADS, 0, stream>>>(x, qout, sout);
}